// MaskingModule_9010841387362
// MI455X (gfx1250) — compile-verified
//
#include <hip/hip_runtime.h>
#include <hip/hip_bf16.h>
#include <stdint.h>

// MaskingModule for MI455X (gfx1250, wave32).
// Memory-bound: ~145 MB total traffic -> ~6us floor @ 23.3 TB/s.
// Uses CDNA5 async-tensor path (global_load_async_to_lds_b128 /
// global_store_async_from_lds_b128 + s_wait_asynccnt) for the gather.

#define N_IMG   32
#define L_SEQ   1024
#define D_DIM   768
#define KEEP    256          // L * (1 - 0.75)
#define NBINS   10

typedef __attribute__((address_space(3))) float4 lds_f4;

// ---------- wave32 helpers ----------
__device__ __forceinline__ unsigned long long shfl_xor_u64(unsigned long long v, int m) {
    unsigned lo = (unsigned)v;
    unsigned hi = (unsigned)(v >> 32);
    lo = __shfl_xor(lo, m, 32);
    hi = __shfl_xor(hi, m, 32);
    return ((unsigned long long)hi << 32) | lo;
}

// =====================================================================
// Kernel 1: per-row entropy.  One wave32 per row of 768 floats.
// =====================================================================
__global__ __launch_bounds__(256) void entropy_kernel(const float* __restrict__ x,
                                                      float* __restrict__ ent) {
    const int row  = (int)((blockIdx.x * blockDim.x + threadIdx.x) >> 5);
    const int lane = (int)(threadIdx.x & 31);
    if (row >= N_IMG * L_SEQ) return;

    const float4* rp = (const float4*)(x + (size_t)row * D_DIM);

    // 24 values per lane via 6 coalesced B128 loads
    float vals[24];
#pragma unroll
    for (int k = 0; k < 6; ++k) {
        float4 v = rp[lane + 32 * k];
        vals[4 * k + 0] = v.x;
        vals[4 * k + 1] = v.y;
        vals[4 * k + 2] = v.z;
        vals[4 * k + 3] = v.w;
    }

    float mn = vals[0], mx = vals[0];
#pragma unroll
    for (int i = 1; i < 24; ++i) {
        mn = fminf(mn, vals[i]);
        mx = fmaxf(mx, vals[i]);
    }
    // wave32 allreduce min/max
#pragma unroll
    for (int m = 16; m > 0; m >>= 1) {
        mn = fminf(mn, __shfl_xor(mn, m, 32));
        mx = fmaxf(mx, __shfl_xor(mx, m, 32));
    }

    const float denom = (mx - mn) + 1e-19f;       // matches reference f32 math
    const float scale = 9.0f / denom;             // (num_bins - 1) / denom

    // branchless histogram: 10 bins x 6 bits packed into one u64
    // (per-lane count <= 24 < 64, no overflow)
    unsigned long long acc = 0ull;
#pragma unroll
    for (int i = 0; i < 24; ++i) {
        int q = (int)((vals[i] - mn) * scale);    // trunc == floor for >= 0
        q = q < 0 ? 0 : (q > 9 ? 9 : q);
        acc += 1ull << (q * 6);
    }

    // repack to 12-bit fields (wave total per bin <= 768 < 4096) and reduce
    unsigned long long lo = 0ull, hi = 0ull;
#pragma unroll
    for (int b = 0; b < 5; ++b) {
        lo |= ((acc >> (6 * b)) & 63ull) << (12 * b);
        hi |= ((acc >> (6 * (b + 5))) & 63ull) << (12 * b);
    }
#pragma unroll
    for (int m = 16; m > 0; m >>= 1) {
        lo += shfl_xor_u64(lo, m);
        hi += shfl_xor_u64(hi, m);
    }

    // lanes 0..9 each handle one bin's  p * log(p + 1e-9)
    float t = 0.0f;
    if (lane < NBINS) {
        unsigned long long src = (lane < 5) ? lo : hi;
        int sh = 12 * (lane % 5);
        float c = (float)((src >> sh) & 4095ull);
        float p = c * (1.0f / 768.0f);
        t = p * __logf(p + 1e-9f);                // p==0 -> 0 (matches ref)
    }
#pragma unroll
    for (int m = 16; m > 0; m >>= 1) t += __shfl_xor(t, m, 32);

    if (lane == 0) ent[row] = -t;
}

// =====================================================================
// Kernel 2: per-image bitonic argsort (descending entropy, stable).
// One block per image; 1024 (key, idx) pairs in LDS.
// =====================================================================
__global__ __launch_bounds__(256) void sort_kernel(const float* __restrict__ ent,
                                                   float* __restrict__ mask,
                                                   float* __restrict__ ids_restore,
                                                   int* __restrict__ keep) {
    __shared__ float k_s[L_SEQ];
    __shared__ int   i_s[L_SEQ];

    const int n = (int)blockIdx.x;
    for (int i = (int)threadIdx.x; i < L_SEQ; i += (int)blockDim.x) {
        k_s[i] = ent[n * L_SEQ + i];
        i_s[i] = i;
    }
    __syncthreads();

    // final order: key descending, index ascending on ties (stable argsort of -e)
    for (int k = 2; k <= L_SEQ; k <<= 1) {
        for (int j = k >> 1; j > 0; j >>= 1) {
            for (int t = (int)threadIdx.x; t < L_SEQ; t += (int)blockDim.x) {
                int ixj = t ^ j;
                if (ixj > t) {
                    bool dir = ((t & k) == 0);    // true: this block in final order
                    float ka = k_s[t], kb = k_s[ixj];
                    int   ia = i_s[t], ib = i_s[ixj];
                    // element at t sorts AFTER element at ixj in final order?
                    bool a_after_b = (ka < kb) || (ka == kb && ia > ib);
                    if (a_after_b == dir) {
                        k_s[t] = kb; k_s[ixj] = ka;
                        i_s[t] = ib; i_s[ixj] = ia;
                    }
                }
            }
            __syncthreads();
        }
    }

    for (int jpos = (int)threadIdx.x; jpos < L_SEQ; jpos += (int)blockDim.x) {
        int orig = i_s[jpos];                     // ids_shuffle[jpos]
        ids_restore[n * L_SEQ + orig] = (float)jpos;
        mask[n * L_SEQ + orig] = (jpos >= KEEP) ? 1.0f : 0.0f;
        if (jpos < KEEP) keep[n * KEEP + jpos] = orig;
    }
}

// =====================================================================
// Kernel 3: gather kept rows via CDNA5 async LDS path.
// One block (192 lanes) per output row; 16 B per lane, global->LDS->global.
// =====================================================================
__global__ __launch_bounds__(192) void gather_kernel(const float* __restrict__ x,
                                                     const int* __restrict__ keep,
                                                     float* __restrict__ xm) {
    __shared__ __align__(16) float4 tile[D_DIM / 4];   // 3 KB

    const int b = (int)blockIdx.x;                 // 0 .. N_IMG*KEEP-1
    const int n = b >> 8;
    const int j = b & (KEEP - 1);
    const int t = (int)threadIdx.x;                // 0..191

    const int src_row = keep[n * KEEP + j];
    const float* src = x  + ((size_t)n * L_SEQ + src_row) * D_DIM + t * 4;
    float*       dst = xm + (size_t)b * D_DIM + t * 4;

    lds_f4* lp = (lds_f4*)tile + t;                // 32-bit LDS address in a VGPR

    // async global -> LDS (ASYNCcnt), wait, async LDS -> global, wait
    asm volatile("global_load_async_to_lds_b128 %0, %1, off"
                 :: "v"(lp), "v"(src) : "memory");
    asm volatile("s_wait_asynccnt 0" ::: "memory");
    asm volatile("global_store_async_from_lds_b128 %0, %1, off"
                 :: "v"(dst), "v"(lp) : "memory");
    asm volatile("s_wait_asynccnt 0" ::: "memory");
}

// =====================================================================
extern "C" void kernel_launch(void* const* d_in, const int* in_sizes, int n_in,
                              void* d_out, int out_size, void* d_ws, size_t ws_size,
                              hipStream_t stream) {
    (void)in_sizes; (void)n_in; (void)out_size; (void)ws_size;

    const float* x = (const float*)d_in[0];

    // outputs, concatenated flat in return order
    float* xm   = (float*)d_out;                               // 32*256*768
    float* mask = xm   + (size_t)N_IMG * KEEP * D_DIM;         // 32*1024
    float* idsr = mask + (size_t)N_IMG * L_SEQ;                // 32*1024

    // scratch: entropies (128 KB) + keep ids (32 KB)
    float* ent  = (float*)d_ws;
    int*   keep = (int*)((char*)d_ws + sizeof(float) * (size_t)N_IMG * L_SEQ);

    // 32K rows, 8 waves (one row each) per 256-thread block
    entropy_kernel<<<(N_IMG * L_SEQ) / 8, 256, 0, stream>>>(x, ent);
    sort_kernel<<<N_IMG, 256, 0, stream>>>(ent, mask, idsr, keep);
    gather_kernel<<<N_IMG * KEEP, 192, 0, stream>>>(x, keep, xm);
}